// VCGAuctioneer_7533372637968
// MI455X (gfx1250) — compile-verified
//
#include <hip/hip_runtime.h>
#include <stdint.h>

#define EXPERTS 128
#define TOPK    8
#define WAVES   8      // waves per block (256 threads, wave32)
#define PIPE    3      // async pipeline depth (distance-2 prefetch)

#if __has_builtin(__builtin_amdgcn_global_load_async_to_lds_b128) && \
    __has_builtin(__builtin_amdgcn_s_wait_asynccnt)
#define USE_ASYNC 1
#else
#define USE_ASYNC 0
#endif

#if defined(__AMDGCN__)
typedef int v4i __attribute__((ext_vector_type(4)));
typedef __attribute__((address_space(1))) v4i* gv4i_ptr;   // global
typedef __attribute__((address_space(3))) v4i* lv4i_ptr;   // LDS
#endif

__device__ __forceinline__ void stage16(const float* __restrict__ g, float* l) {
#if USE_ASYNC
  // async DMA: 32 lanes x 16B = 512B per wave, tracked on ASYNCcnt
  __builtin_amdgcn_global_load_async_to_lds_b128(
      (gv4i_ptr)(uintptr_t)g, (lv4i_ptr)l, /*offset=*/0, /*cpol=*/0);
#else
  *(float4*)l = *(const float4*)g;
#endif
}

__device__ __forceinline__ void wait_async_le(int n) {
#if USE_ASYNC
  if (n == 0)      __builtin_amdgcn_s_wait_asynccnt(0);
  else if (n == 1) __builtin_amdgcn_s_wait_asynccnt(1);
  else             __builtin_amdgcn_s_wait_asynccnt(2);
#else
  (void)n;  // synchronous fallback: data already in LDS
#endif
}

__global__ __launch_bounds__(256, 4)
void vcg_topk_kernel(const float* __restrict__ conf,
                     const float* __restrict__ wealth,
                     float* __restrict__ out_sel,
                     float* __restrict__ out_rw,
                     float* __restrict__ out_pay,
                     int T)
{
  __shared__ __align__(16) float stage[WAVES][PIPE][EXPERTS];

  const int lane = threadIdx.x & 31;
  const int wv   = threadIdx.x >> 5;
  const int nw   = gridDim.x * WAVES;          // total waves
  const int gw   = blockIdx.x * WAVES + wv;    // global wave id

  // per-lane wealth: experts lane*4 .. lane*4+3 (512B total, L2-resident)
  float4 w4 = reinterpret_cast<const float4*>(wealth)[lane];
  float wl[4] = {w4.x, w4.y, w4.z, w4.w};

  // pipeline prologue: prefetch tokens gw and gw+nw
  {
    int t0 = gw, t1 = gw + nw;
    if (t0 < T) stage16(conf + (size_t)t0 * EXPERTS + lane * 4, &stage[wv][0][lane * 4]);
    if (t1 < T) stage16(conf + (size_t)t1 * EXPERTS + lane * 4, &stage[wv][1][lane * 4]);
  }

  int it = 0;
  for (int token = gw; token < T; token += nw, ++it) {
    const int buf = it % PIPE;
    const int t2  = token + 2 * nw;
    if (t2 < T)
      stage16(conf + (size_t)t2 * EXPERTS + lane * 4,
              &stage[wv][(it + 2) % PIPE][lane * 4]);

    // wait until the current buffer's DMA has landed
    if (token + nw < T) { if (t2 < T) wait_async_le(2); else wait_async_le(1); }
    else                wait_async_le(0);

    float4 c4 = *reinterpret_cast<float4*>(&stage[wv][buf][lane * 4]);
    float cl[4] = {c4.x, c4.y, c4.z, c4.w};

    // bids + sortable keys: bids >= 0 so float bits are order-monotonic.
    // key = (bits << 32) | (0xFFFFFFFF - idx)  => max key == (max value, min index)
    unsigned long long key[4];
    float bid[4];
#pragma unroll
    for (int k = 0; k < 4; ++k) {
      float b = cl[k] * wl[k];
      bid[k] = b;
      unsigned e = (unsigned)(lane * 4 + k);
      key[k] = ((unsigned long long)__float_as_uint(b) << 32) | (0xFFFFFFFFu - e);
    }
    unsigned long long lbest = key[0];
#pragma unroll
    for (int k = 1; k < 4; ++k) lbest = (key[k] > lbest) ? key[k] : lbest;

    // iterative top-8 extraction: wave32 butterfly arg-max (5 shuffle steps)
    float tv[TOPK];
    int   ti[TOPK];
#pragma unroll
    for (int j = 0; j < TOPK; ++j) {
      unsigned long long b = lbest;
#pragma unroll
      for (int m = 16; m >= 1; m >>= 1) {
        unsigned long long o = __shfl_xor(b, m, 32);
        b = (o > b) ? o : b;
      }
      tv[j] = __uint_as_float((unsigned)(b >> 32));
      ti[j] = (int)(0xFFFFFFFFu - (unsigned)(b & 0xFFFFFFFFull));
      if (b == lbest) {            // keys are globally unique: exactly one owner
#pragma unroll
        for (int k = 0; k < 4; ++k) if (key[k] == b) key[k] = 0ull;
        lbest = key[0];
#pragma unroll
        for (int k = 1; k < 4; ++k) lbest = (key[k] > lbest) ? key[k] : lbest;
      }
    }

    // softmax denominator over all 128 (max-stable; top1 is the max)
    const float mx = tv[0];
    float s = 0.f;
#pragma unroll
    for (int k = 0; k < 4; ++k) s += expf(bid[k] - mx);
#pragma unroll
    for (int m = 16; m >= 1; m >>= 1) s += __shfl_xor(s, m, 32);

    // routing weights: soft[sel] renormalized with +1e-8.
    // one reciprocal instead of 8 divisions (1-2 ulp vs reference, VALU-bound win)
    const float invS = 1.0f / s;
    float rw[TOPK];
    float rsum = 0.f;
#pragma unroll
    for (int j = 0; j < TOPK; ++j) { rw[j] = expf(tv[j] - mx) * invS; rsum += rw[j]; }
    const float invPden = 1.0f / (rsum + 1e-8f);

    float sum8 = 0.f;
#pragma unroll
    for (int j = 0; j < TOPK; ++j) sum8 += tv[j];

    // VCG payments + per-lane output selection (all static indexing -> registers)
    int   msel = 0;
    float mrw = 0.f, mpay = 0.f;
#pragma unroll
    for (int j = 0; j < TOPK; ++j) {
      float w7 = 0.f;                       // top-7 welfare without winner j
#pragma unroll
      for (int i = 0; i < TOPK; ++i) if (i != j) w7 += tv[i];
      float pay = fmaxf(w7 - (sum8 - tv[j]), 0.f);
      float r   = rw[j] * invPden;
      if (lane == j) { msel = ti[j]; mrw = r; mpay = pay; }
    }

    if (lane < TOPK) {
      size_t o = (size_t)token * TOPK + lane;
      out_sel[o] = (float)msel;
      out_rw[o]  = mrw;
      out_pay[o] = mpay;
    }
  }
}

extern "C" void kernel_launch(void* const* d_in, const int* in_sizes, int n_in,
                              void* d_out, int out_size, void* d_ws, size_t ws_size,
                              hipStream_t stream) {
  (void)n_in; (void)d_ws; (void)ws_size; (void)out_size;
  const float* conf   = (const float*)d_in[0];   // (8, 8192, 128) f32
  const float* wealth = (const float*)d_in[1];   // (128,) f32

  const int T = in_sizes[0] / EXPERTS;           // 65536 tokens
  float* out     = (float*)d_out;                // [selected | routing | payments]
  float* out_sel = out;
  float* out_rw  = out + (size_t)T * TOPK;
  float* out_pay = out + 2 * (size_t)T * TOPK;

  const int blocks = 1024;                       // 8192 waves -> 8 tokens/wave
  hipLaunchKernelGGL(vcg_topk_kernel, dim3(blocks), dim3(256), 0, stream,
                     conf, wealth, out_sel, out_rw, out_pay, T);
}